// FlowPredictorGNN_89988154786542
// MI455X (gfx1250) — compile-verified
//
#include <hip/hip_runtime.h>
#include <math.h>

#define NN 50000
#define NE 1600000
#define HD 64
#define SLOPE 0.2f
#define TPB 256

typedef __attribute__((ext_vector_type(2))) float v2f;
typedef __attribute__((ext_vector_type(8))) float v8f;

__device__ __forceinline__ float lrelu(float x) { return x > 0.f ? x : SLOPE * x; }

// Monotonic order-preserving float -> uint encoding for atomicMax-based segment max.
__device__ __forceinline__ unsigned fenc(float f) {
    unsigned u = __float_as_uint(f);
    return (u & 0x80000000u) ? ~u : (u | 0x80000000u);
}
__device__ __forceinline__ float fdec(unsigned u) {
    return (u & 0x80000000u) ? __uint_as_float(u & 0x7fffffffu) : __uint_as_float(~u);
}

// ---------- layer linear transforms ----------

// Layer 1: h[i][j] = x[i] * W1[j]   (K = 1 outer product)
__global__ void k_lin1(const float* __restrict__ x, const float* __restrict__ W1,
                       float* __restrict__ h) {
    int t = blockIdx.x * blockDim.x + threadIdx.x;
    if (t < NN * HD) h[t] = x[t >> 6] * W1[t & 63];
}

// Layer 2: C[NN,64] = A[NN,64] x B[64,64] via V_WMMA_F32_16X16X4_F32.
// One wave per 16x16 C tile; 4 waves/block cover the 64 output columns of a
// 16-row strip. NN == 3125*16 exactly -> no bounds checks.
__global__ void __launch_bounds__(128)
k_gemm64_wmma(const float* __restrict__ A, const float* __restrict__ B,
              float* __restrict__ C) {
    int lane    = threadIdx.x & 31;
    int wave    = threadIdx.x >> 5;
    int rowBase = blockIdx.x * 16;
    int colBase = wave * 16;
    int m       = lane & 15;            // A-matrix row within tile
    int khalf   = (lane >> 4) << 1;     // lanes 16..31 hold K+2,K+3
    int n       = colBase + (lane & 15);// B/C column

    v8f acc = {};
#pragma unroll
    for (int k0 = 0; k0 < HD; k0 += 4) {
        v2f a, b;
        const float* arow = A + (rowBase + m) * HD + k0 + khalf;
        a.x = arow[0];
        a.y = arow[1];
        b.x = B[(k0 + khalf) * HD + n];
        b.y = B[(k0 + khalf + 1) * HD + n];
        acc = __builtin_amdgcn_wmma_f32_16x16x4_f32(
            /*neg_a=*/false, a, /*neg_b=*/false, b,
            /*c_mod=*/(short)0, acc, /*reuse_a=*/false, /*reuse_b=*/false);
    }
    // C/D layout: VGPR r = row r (lanes 0-15) / row r+8 (lanes 16-31)
    int rOff = (lane >> 4) * 8;
#pragma unroll
    for (int r = 0; r < 8; ++r)
        C[(rowBase + rOff + r) * HD + colBase + (lane & 15)] = acc[r];
}

// Layer 3 linear (64 -> 1) fused with its attention scores.
__global__ void k_lin3_scores(const float* __restrict__ h2, const float* __restrict__ w3,
                              const float* __restrict__ a_s, const float* __restrict__ a_d,
                              float* __restrict__ h3, float* __restrict__ ss,
                              float* __restrict__ sd) {
    int i = blockIdx.x * blockDim.x + threadIdx.x;
    if (i >= NN) return;
    float v = 0.f;
#pragma unroll
    for (int j = 0; j < HD; ++j) v += h2[i * HD + j] * w3[j];
    h3[i] = v;
    ss[i] = v * a_s[0];
    sd[i] = v * a_d[0];
}

// ---------- attention scores (per node, 64-dim) ----------
__global__ void k_scores(const float* __restrict__ h, const float* __restrict__ av_s,
                         const float* __restrict__ av_d, float* __restrict__ ss,
                         float* __restrict__ sd) {
    int i = blockIdx.x * blockDim.x + threadIdx.x;
    if (i >= NN) return;
    float a = 0.f, b = 0.f;
#pragma unroll
    for (int j = 0; j < HD; ++j) {
        float v = h[i * HD + j];
        a += v * av_s[j];
        b += v * av_d[j];
    }
    ss[i] = a;
    sd[i] = b;
}

// ---------- init ----------
__global__ void k_zero(float* __restrict__ p, int n) {
    int i = blockIdx.x * blockDim.x + threadIdx.x;
    if (i < n) p[i] = 0.f;
}
__global__ void k_init_seg(unsigned* __restrict__ mx, float* __restrict__ dn) {
    int i = blockIdx.x * blockDim.x + threadIdx.x;
    if (i < NN) { mx[i] = 0u; dn[i] = 0.f; }  // 0u < fenc(-inf): safe floor
}

// ---------- segment softmax edge passes (E real edges + N self-loops) ----------
__global__ void k_edge_max(const int* __restrict__ src, const int* __restrict__ dst,
                           const float* __restrict__ ss, const float* __restrict__ sd,
                           unsigned* __restrict__ mx) {
    int t = blockIdx.x * blockDim.x + threadIdx.x;
    if (t >= NE + NN) return;
    int s, d;
    if (t < NE) { s = src[t]; d = dst[t]; } else { s = d = t - NE; }
    atomicMax(&mx[d], fenc(lrelu(ss[s] + sd[d])));
}

__global__ void k_edge_denom(const int* __restrict__ src, const int* __restrict__ dst,
                             const float* __restrict__ ss, const float* __restrict__ sd,
                             const unsigned* __restrict__ mx, float* __restrict__ dn) {
    int t = blockIdx.x * blockDim.x + threadIdx.x;
    if (t >= NE + NN) return;
    int s, d;
    if (t < NE) { s = src[t]; d = dst[t]; } else { s = d = t - NE; }
    float e = lrelu(ss[s] + sd[d]);
    atomicAdd(&dn[d], expf(e - fdec(mx[d])));
}

// alpha-weighted scatter, 64-dim features: one wave32 per edge, lane = dim.
// Gather of h[src] and the 64 atomicAdds are fully coalesced 256B transactions.
__global__ void k_edge_aggr64(const int* __restrict__ src, const int* __restrict__ dst,
                              const float* __restrict__ ss, const float* __restrict__ sd,
                              const unsigned* __restrict__ mx, const float* __restrict__ dn,
                              const float* __restrict__ hlin, float* __restrict__ aggr) {
    int t = blockIdx.x * blockDim.x + threadIdx.x;
    int e = t >> 5, lane = t & 31;
    if (e >= NE + NN) return;
    int s, d;
    if (e < NE) { s = src[e]; d = dst[e]; } else { s = d = e - NE; }
    float ev    = lrelu(ss[s] + sd[d]);
    float alpha = expf(ev - fdec(mx[d])) / dn[d];
    const float* hs = hlin + s * HD;
    float* ad       = aggr + d * HD;
    atomicAdd(&ad[lane],      alpha * hs[lane]);
    atomicAdd(&ad[lane + 32], alpha * hs[lane + 32]);
}

// scalar-feature variant for layer 3
__global__ void k_edge_aggr1(const int* __restrict__ src, const int* __restrict__ dst,
                             const float* __restrict__ ss, const float* __restrict__ sd,
                             const unsigned* __restrict__ mx, const float* __restrict__ dn,
                             const float* __restrict__ h3, float* __restrict__ ag) {
    int t = blockIdx.x * blockDim.x + threadIdx.x;
    if (t >= NE + NN) return;
    int s, d;
    if (t < NE) { s = src[t]; d = dst[t]; } else { s = d = t - NE; }
    float ev    = lrelu(ss[s] + sd[d]);
    float alpha = expf(ev - fdec(mx[d])) / dn[d];
    atomicAdd(&ag[d], alpha * h3[s]);
}

// ---------- epilogues ----------
__global__ void k_finish_relu(float* __restrict__ h, const float* __restrict__ b) {
    int t = blockIdx.x * blockDim.x + threadIdx.x;
    if (t < NN * HD) {
        float v = h[t] + b[t & 63];
        h[t] = v > 0.f ? v : 0.f;
    }
}
__global__ void k_finish_out(const float* __restrict__ ag, const float* __restrict__ b3,
                             float* __restrict__ out) {
    int i = blockIdx.x * blockDim.x + threadIdx.x;
    if (i < NN) out[i] = ag[i] + b3[0];
}

// ---------- host ----------
extern "C" void kernel_launch(void* const* d_in, const int* in_sizes, int n_in,
                              void* d_out, int out_size, void* d_ws, size_t ws_size,
                              hipStream_t stream) {
    (void)in_sizes; (void)n_in; (void)out_size; (void)ws_size;
    const float* x   = (const float*)d_in[0];
    const int*   ei  = (const int*)d_in[1];   // [2, E] row-major
    const int*   src = ei;
    const int*   dst = ei + NE;
    const float* W1  = (const float*)d_in[3];
    const float* as1 = (const float*)d_in[4];
    const float* ad1 = (const float*)d_in[5];
    const float* b1  = (const float*)d_in[6];
    const float* W2  = (const float*)d_in[7];
    const float* as2 = (const float*)d_in[8];
    const float* ad2 = (const float*)d_in[9];
    const float* b2  = (const float*)d_in[10];
    const float* W3  = (const float*)d_in[11];
    const float* as3 = (const float*)d_in[12];
    const float* ad3 = (const float*)d_in[13];
    const float* b3  = (const float*)d_in[14];
    float* out = (float*)d_out;

    // workspace carve-up (~27 MB)
    float*    h1  = (float*)d_ws;
    float*    h2  = h1 + (size_t)NN * HD;
    float*    ss  = h2 + (size_t)NN * HD;
    float*    sd  = ss + NN;
    unsigned* mx  = (unsigned*)(sd + NN);
    float*    dn  = (float*)(mx + NN);
    float*    h3  = dn + NN;
    float*    ag3 = h3 + NN;

    const int EV  = NE + NN;                       // virtual edges (incl. self-loops)
    const int gN  = (NN + TPB - 1) / TPB;
    const int gNH = (NN * HD + TPB - 1) / TPB;
    const int gE  = (EV + TPB - 1) / TPB;
    const int gEw = (EV + (TPB / 32) - 1) / (TPB / 32);  // one wave per edge

    // ---- Layer 1 ----
    k_lin1<<<gNH, TPB, 0, stream>>>(x, W1, h1);
    k_scores<<<gN, TPB, 0, stream>>>(h1, as1, ad1, ss, sd);
    k_init_seg<<<gN, TPB, 0, stream>>>(mx, dn);
    k_zero<<<gNH, TPB, 0, stream>>>(h2, NN * HD);
    k_edge_max<<<gE, TPB, 0, stream>>>(src, dst, ss, sd, mx);
    k_edge_denom<<<gE, TPB, 0, stream>>>(src, dst, ss, sd, mx, dn);
    k_edge_aggr64<<<gEw, TPB, 0, stream>>>(src, dst, ss, sd, mx, dn, h1, h2);
    k_finish_relu<<<gNH, TPB, 0, stream>>>(h2, b1);

    // ---- Layer 2 (WMMA GEMM) ----
    k_gemm64_wmma<<<NN / 16, 128, 0, stream>>>(h2, W2, h1);
    k_scores<<<gN, TPB, 0, stream>>>(h1, as2, ad2, ss, sd);
    k_init_seg<<<gN, TPB, 0, stream>>>(mx, dn);
    k_zero<<<gNH, TPB, 0, stream>>>(h2, NN * HD);
    k_edge_max<<<gE, TPB, 0, stream>>>(src, dst, ss, sd, mx);
    k_edge_denom<<<gE, TPB, 0, stream>>>(src, dst, ss, sd, mx, dn);
    k_edge_aggr64<<<gEw, TPB, 0, stream>>>(src, dst, ss, sd, mx, dn, h1, h2);
    k_finish_relu<<<gNH, TPB, 0, stream>>>(h2, b2);

    // ---- Layer 3 ----
    k_lin3_scores<<<gN, TPB, 0, stream>>>(h2, W3, as3, ad3, h3, ss, sd);
    k_init_seg<<<gN, TPB, 0, stream>>>(mx, dn);
    k_zero<<<gN, TPB, 0, stream>>>(ag3, NN);
    k_edge_max<<<gE, TPB, 0, stream>>>(src, dst, ss, sd, mx);
    k_edge_denom<<<gE, TPB, 0, stream>>>(src, dst, ss, sd, mx, dn);
    k_edge_aggr1<<<gE, TPB, 0, stream>>>(src, dst, ss, sd, mx, dn, h3, ag3);
    k_finish_out<<<gN, TPB, 0, stream>>>(ag3, b3, out);
}